// GAT_11819749999222
// MI455X (gfx1250) — compile-verified
//
#include <hip/hip_runtime.h>
#include <hip/hip_bf16.h>

// GAT layer: N=100000 nodes, E=1.6M edges, IN=256, H=4, C=64.
// Phase 1: xp = x@W via v_wmma_f32_16x16x4_f32, with the per-block W slice
//          staged into LDS through the gfx1250 async-copy path
//          (global_load_async_to_lds_b128 + s_wait_asynccnt).
// Phase 2: per-node attention scores (one wave per node).
// Phase 3: segment max (uint-key atomicMax), segment sum (atomicAdd).
// Phase 4: weighted scatter-add, one wave32 per edge, 8 coalesced sweeps.
// Phase 5: elu(agg + bias) in place on d_out.

typedef __attribute__((ext_vector_type(2))) float v2f;
typedef __attribute__((ext_vector_type(8))) float v8f;

#define HC 256   // H*C
#define KDIM 256 // IN

// ---------------- order-preserving float<->uint key for atomic max ---------
__device__ __forceinline__ unsigned fkey(float f) {
  unsigned u = __float_as_uint(f);
  return (u & 0x80000000u) ? ~u : (u | 0x80000000u);
}
__device__ __forceinline__ float funkey(unsigned k) {
  return (k & 0x80000000u) ? __uint_as_float(k & 0x7fffffffu)
                           : __uint_as_float(~k);
}

// ---------------- Phase 1: xp = x @ W  (FP32 WMMA, LDS-staged B) -----------
// Block = 256 threads = 8 waves; block covers rows [128*bx, 128*bx+128) x
// cols [64*by, 64*by+64). W slice (256x64 f32 = 64KB) is async-copied to LDS.
__global__ void __launch_bounds__(256) gat_gemm_wmma(
    const float* __restrict__ x, const float* __restrict__ Wm,
    float* __restrict__ xp, int N) {
  __shared__ float sW[KDIM * 64];                  // 64 KB, row-major [256][64]

  const int tid  = threadIdx.x;
  const int lane = tid & 31;
  const int wave = tid >> 5;
  const int col0 = blockIdx.y * 64;

  // --- cooperative async copy: W[0:256, col0:col0+64] -> sW ---------------
  {
    const unsigned ldsbase = (unsigned)(uintptr_t)(void*)sW; // low 32 bits = LDS offset
#pragma unroll
    for (int i = 0; i < 16; ++i) {                 // 16 rounds x 256 lanes x 16B
      const int q   = i * 256 + tid;               // 16-byte chunk index 0..4095
      const int row = q >> 4;                      // 0..255
      const int cq  = (q & 15) << 2;               // col offset 0,4,...,60
      const float* gsrc = Wm + (size_t)row * HC + col0 + cq;
      const unsigned ldst = ldsbase + ((unsigned)(row * 64 + cq) << 2);
      asm volatile("global_load_async_to_lds_b128 %0, %1, off"
                   :: "v"(ldst), "v"(gsrc) : "memory");
    }
    asm volatile("s_wait_asynccnt 0" ::: "memory");
  }
  __syncthreads();

  const int rowTile = blockIdx.x * 8 + wave;
  const int row0 = rowTile * 16;
  if (row0 < N) {                                  // wave-uniform predicate
    const int mrow = row0 + (lane & 15);           // A: M striped over lanes 0-15
    const int koff = (lane >> 4) * 2;              // lanes 16-31 carry K+2,K+3
    const int c    = lane & 15;                    // B/C/D: N striped over lanes
    const float* __restrict__ xrow = x + (size_t)mrow * KDIM;

    v8f acc0 = {}, acc1 = {}, acc2 = {}, acc3 = {};

    for (int k = 0; k < KDIM; k += 4) {
      // A tile 16x4: per-lane float2 (K=koff, koff+1), 8B aligned.
      v2f a = *(const v2f*)(xrow + k + koff);
      // B tile 4x16 per 16-col slice from LDS:
      //   b.x = W[k+koff][col0+c+16j], b.y = W[k+koff+1][col0+c+16j]
      const float* bp = &sW[(k + koff) * 64 + c];
      v2f b0 = { bp[0],  bp[64] };
      v2f b1 = { bp[16], bp[80] };
      v2f b2 = { bp[32], bp[96] };
      v2f b3 = { bp[48], bp[112] };
      acc0 = __builtin_amdgcn_wmma_f32_16x16x4_f32(false, a, false, b0, (short)0, acc0, false, false);
      acc1 = __builtin_amdgcn_wmma_f32_16x16x4_f32(false, a, false, b1, (short)0, acc1, false, false);
      acc2 = __builtin_amdgcn_wmma_f32_16x16x4_f32(false, a, false, b2, (short)0, acc2, false, false);
      acc3 = __builtin_amdgcn_wmma_f32_16x16x4_f32(false, a, false, b3, (short)0, acc3, false, false);
    }

    // C/D layout: VGPR v holds M=v (lanes 0-15), M=v+8 (lanes 16-31), N=lane&15
    const int rbase = row0 + ((lane >> 4) << 3);
#pragma unroll
    for (int v = 0; v < 8; ++v) {
      float* orow = xp + (size_t)(rbase + v) * HC + col0 + c;
      orow[0]  = acc0[v];
      orow[16] = acc1[v];
      orow[32] = acc2[v];
      orow[48] = acc3[v];
    }
  }
}

// ---------------- Phase 2: a_src/a_dst per node ----------------------------
// One wave per node; lane L reduces channels [8L, 8L+8) (head h = L>>3).
__global__ void __launch_bounds__(256) gat_scores(
    const float* __restrict__ xp, const float* __restrict__ att_src,
    const float* __restrict__ att_dst, float* __restrict__ a_src,
    float* __restrict__ a_dst, int N) {
  const int lane = threadIdx.x & 31;
  const int node = blockIdx.x * 8 + (threadIdx.x >> 5);
  if (node >= N) return;
  const float* row = xp + (size_t)node * HC;
  const int base = lane * 8;
  float s = 0.f, d = 0.f;
#pragma unroll
  for (int j = 0; j < 8; ++j) {
    float v = row[base + j];
    s += v * att_src[base + j];
    d += v * att_dst[base + j];
  }
#pragma unroll
  for (int off = 4; off >= 1; off >>= 1) {
    s += __shfl_xor(s, off, 32);
    d += __shfl_xor(d, off, 32);
  }
  if ((lane & 7) == 0) {
    const int h = lane >> 3;
    a_src[node * 4 + h] = s;
    a_dst[node * 4 + h] = d;
  }
}

// ---------------- Phase 3a: segment max of leaky_relu logits ---------------
__global__ void gat_edge_max(const int* __restrict__ srcI,
                             const int* __restrict__ dstI,
                             const float* __restrict__ a_src,
                             const float* __restrict__ a_dst,
                             unsigned* __restrict__ umax,
                             long long E, long long ET) {
  long long e = (long long)blockIdx.x * blockDim.x + threadIdx.x;
  if (e >= ET) return;
  int s, d;
  if (e < E) { s = srcI[e]; d = dstI[e]; } else { s = d = (int)(e - E); }
#pragma unroll
  for (int h = 0; h < 4; ++h) {
    float v = a_src[(size_t)s * 4 + h] + a_dst[(size_t)d * 4 + h];
    v = v > 0.f ? v : 0.2f * v;                      // LeakyReLU(0.2)
    atomicMax(&umax[(size_t)d * 4 + h], fkey(v));
  }
}

// ---------------- Phase 3b: segment sum of exp(alpha - max) ----------------
__global__ void gat_edge_denom(const int* __restrict__ srcI,
                               const int* __restrict__ dstI,
                               const float* __restrict__ a_src,
                               const float* __restrict__ a_dst,
                               const unsigned* __restrict__ umax,
                               float* __restrict__ denom,
                               long long E, long long ET) {
  long long e = (long long)blockIdx.x * blockDim.x + threadIdx.x;
  if (e >= ET) return;
  int s, d;
  if (e < E) { s = srcI[e]; d = dstI[e]; } else { s = d = (int)(e - E); }
#pragma unroll
  for (int h = 0; h < 4; ++h) {
    float v = a_src[(size_t)s * 4 + h] + a_dst[(size_t)d * 4 + h];
    v = v > 0.f ? v : 0.2f * v;
    float ex = expf(v - funkey(umax[(size_t)d * 4 + h]));
    atomicAdd(&denom[(size_t)d * 4 + h], ex);
  }
}

// ---------------- Phase 4: weighted scatter-add ----------------------------
// One wave32 per edge: 8 coalesced 128B sweeps over the 256 channels.
__global__ void __launch_bounds__(256) gat_scatter(
    const int* __restrict__ srcI, const int* __restrict__ dstI,
    const float* __restrict__ xp, const float* __restrict__ a_src,
    const float* __restrict__ a_dst, const unsigned* __restrict__ umax,
    const float* __restrict__ denom, float* __restrict__ agg,
    long long E, long long ET) {
  const int lane = threadIdx.x & 31;
  long long e = (long long)blockIdx.x * 8 + (threadIdx.x >> 5);
  if (e >= ET) return;
  int s, d;
  if (e < E) { s = srcI[e]; d = dstI[e]; } else { s = d = (int)(e - E); }

  float w[4];
#pragma unroll
  for (int h = 0; h < 4; ++h) {
    float v = a_src[(size_t)s * 4 + h] + a_dst[(size_t)d * 4 + h];
    v = v > 0.f ? v : 0.2f * v;
    w[h] = expf(v - funkey(umax[(size_t)d * 4 + h])) /
           denom[(size_t)d * 4 + h];
  }
  const float* xr = xp + (size_t)s * HC;
  float* ar = agg + (size_t)d * HC;
#pragma unroll
  for (int j = 0; j < 8; ++j) {             // element idx = j*32+lane, head = j>>1
    const int idx = j * 32 + lane;
    atomicAdd(&ar[idx], xr[idx] * w[j >> 1]);
  }
}

// ---------------- Phase 5: out = elu(agg + bias) in place ------------------
__global__ void gat_finalize(float* __restrict__ out,
                             const float* __restrict__ bias, long long total) {
  long long i = (long long)blockIdx.x * blockDim.x + threadIdx.x;
  if (i >= total) return;
  float v = out[i] + bias[i & (HC - 1)];
  out[i] = v > 0.f ? v : expm1f(v);
}

extern "C" void kernel_launch(void* const* d_in, const int* in_sizes, int n_in,
                              void* d_out, int out_size, void* d_ws, size_t ws_size,
                              hipStream_t stream) {
  const float* x       = (const float*)d_in[0];
  const int*   ei      = (const int*)d_in[1];   // edge_index as int32 per harness
  const float* Wm      = (const float*)d_in[2];
  const float* att_src = (const float*)d_in[3];
  const float* att_dst = (const float*)d_in[4];
  const float* bias    = (const float*)d_in[5];

  const long long N  = in_sizes[0] / HC;
  const long long E  = in_sizes[1] / 2;
  const long long ET = E + N;
  const int* srcI = ei;
  const int* dstI = ei + E;

  // Workspace layout
  char* p = (char*)d_ws;
  float*    xp    = (float*)p;    p += (size_t)N * HC * sizeof(float); // 102.4 MB
  float*    a_src = (float*)p;    p += (size_t)N * 4 * sizeof(float);
  float*    a_dst = (float*)p;    p += (size_t)N * 4 * sizeof(float);
  unsigned* umax  = (unsigned*)p; p += (size_t)N * 4 * sizeof(unsigned);
  float*    denom = (float*)p;    p += (size_t)N * 4 * sizeof(float);
  float*    agg   = (float*)d_out;                 // accumulate directly in d_out

  // 0 < key(-inf)=0x007fffff: memset-0 valid init for the uint max keys
  hipMemsetAsync(umax, 0, (size_t)N * 4 * sizeof(unsigned), stream);
  hipMemsetAsync(denom, 0, (size_t)N * 4 * sizeof(float), stream);
  hipMemsetAsync(d_out, 0, (size_t)out_size * sizeof(float), stream);

  dim3 ggrid((unsigned)((N + 127) / 128), 4, 1);
  gat_gemm_wmma<<<ggrid, 256, 0, stream>>>(x, Wm, xp, (int)N);
  gat_scores<<<(int)((N + 7) / 8), 256, 0, stream>>>(xp, att_src, att_dst,
                                                     a_src, a_dst, (int)N);
  gat_edge_max<<<(int)((ET + 255) / 256), 256, 0, stream>>>(srcI, dstI, a_src,
                                                            a_dst, umax, E, ET);
  gat_edge_denom<<<(int)((ET + 255) / 256), 256, 0, stream>>>(
      srcI, dstI, a_src, a_dst, umax, denom, E, ET);
  gat_scatter<<<(int)((ET + 7) / 8), 256, 0, stream>>>(
      srcI, dstI, xp, a_src, a_dst, umax, denom, agg, E, ET);
  gat_finalize<<<(int)((N * HC + 255) / 256), 256, 0, stream>>>(agg, bias,
                                                                N * HC);
}